// ClusteringModel_8864812498972
// MI455X (gfx1250) — compile-verified
//
#include <hip/hip_runtime.h>

typedef float v2f __attribute__((ext_vector_type(2)));
typedef float v4f __attribute__((ext_vector_type(4)));
typedef float v8f __attribute__((ext_vector_type(8)));

#define NROWS 131072
#define DIM   784
#define KC    64
#define LDSTR 68   // LDS row stride (floats): bank-conflict-free for store + b64 reload

// ---------------- prep: centersT[d][k] = centers[k][d]; cn2[k] = sum_d centers[k][d]^2
__global__ __launch_bounds__(256) void prep_kernel(const float* __restrict__ centers,
                                                   float* __restrict__ centersT,
                                                   float* __restrict__ cn2) {
    __shared__ float red[256];
    const int k = blockIdx.x;           // 0..63
    float partial = 0.0f;
    for (int d = threadIdx.x; d < DIM; d += 256) {
        float v = centers[k * DIM + d];
        centersT[d * KC + k] = v;
        partial += v * v;
    }
    red[threadIdx.x] = partial;
    __syncthreads();
    for (int s = 128; s > 0; s >>= 1) {
        if (threadIdx.x < s) red[threadIdx.x] += red[threadIdx.x + s];
        __syncthreads();
    }
    if (threadIdx.x == 0) cn2[k] = red[0];
}

// ---------------- fused: logits (WMMA f32) -> softmax -> reconstruction (WMMA f32) + x0 copy
__global__ __launch_bounds__(256) void cluster_kernel(const float* __restrict__ x,
                                                      const float* __restrict__ centers,
                                                      const float* __restrict__ centersT,
                                                      const float* __restrict__ cn2,
                                                      float* __restrict__ xrec,
                                                      float* __restrict__ xcopy) {
    __shared__ float lds_w[8 * 16 * LDSTR];   // per-wave 16x64 W tile (padded)

    const int lane = threadIdx.x & 31;
    const int wave = threadIdx.x >> 5;
    const int half = lane >> 4;               // 0: lanes 0-15, 1: lanes 16-31
    const int ln   = lane & 15;
    const int koff = half * 2;                // A/B frag K offset per ISA 16x16x4 f32 layout
    const size_t row0 = (size_t)blockIdx.x * 128 + (size_t)wave * 16;

    // ================= GEMM1: logits[16 x 64] = x_tile[16 x 784] * centers^T =================
    v8f zero = {};
    v8f acc[4];
    #pragma unroll
    for (int t = 0; t < 4; ++t) acc[t] = zero;

    const float* xrow = x + (row0 + (size_t)ln) * DIM;
    #pragma unroll 4
    for (int k0 = 0; k0 < DIM; k0 += 4) {
        v2f a = *(const v2f*)(xrow + k0 + koff);           // A: M=ln, K pair (k0+koff, +1)
        #pragma unroll
        for (int t = 0; t < 4; ++t) {
            // B[k][n] = centers[n][k]: pair along k is contiguous in centers row n
            v2f b = *(const v2f*)(centers + (size_t)(t * 16 + ln) * DIM + k0 + koff);
            acc[t] = __builtin_amdgcn_wmma_f32_16x16x4_f32(false, a, false, b,
                                                           (short)0, acc[t], false, false);
        }
    }

    // ================= softmax over K=64 (C layout: element (v,lane) = [M=v+8*half][N=16t+ln])
    #pragma unroll
    for (int t = 0; t < 4; ++t) {
        float bias = 10.0f * cn2[t * 16 + ln];             // 0.5*SCALE*||c||^2
        acc[t] = acc[t] * 20.0f - bias;                    // logits
    }
    v8f rm = acc[0];
    #pragma unroll
    for (int t = 1; t < 4; ++t)
        #pragma unroll
        for (int v = 0; v < 8; ++v) rm[v] = fmaxf(rm[v], acc[t][v]);
    #pragma unroll
    for (int v = 0; v < 8; ++v) {
        float m = rm[v];
        m = fmaxf(m, __shfl_xor(m, 8, 32));
        m = fmaxf(m, __shfl_xor(m, 4, 32));
        m = fmaxf(m, __shfl_xor(m, 2, 32));
        m = fmaxf(m, __shfl_xor(m, 1, 32));
        rm[v] = m;
    }
    v8f rs = zero;
    #pragma unroll
    for (int t = 0; t < 4; ++t) {
        #pragma unroll
        for (int v = 0; v < 8; ++v) acc[t][v] = __expf(acc[t][v] - rm[v]);
        rs += acc[t];
    }
    #pragma unroll
    for (int v = 0; v < 8; ++v) {
        float s = rs[v];
        s += __shfl_xor(s, 8, 32);
        s += __shfl_xor(s, 4, 32);
        s += __shfl_xor(s, 2, 32);
        s += __shfl_xor(s, 1, 32);
        rs[v] = 1.0f / s;
    }
    #pragma unroll
    for (int t = 0; t < 4; ++t) acc[t] *= rs;              // w = softmax weights

    // spill W[16x64] to LDS in row-major [m][n] (per-wave region; LDS pipe is in-order per wave)
    float* wbase = lds_w + wave * 16 * LDSTR;
    #pragma unroll
    for (int t = 0; t < 4; ++t)
        #pragma unroll
        for (int v = 0; v < 8; ++v)
            wbase[(v + 8 * half) * LDSTR + t * 16 + ln] = acc[t][v];

    // ================= GEMM2: xrec[16 x 784] = W[16 x 64] * centers[64 x 784] =================
    // Whole A matrix (16x64) as 16 frags, reused across all 49 output tiles.
    v2f af[16];
    const float* arow = wbase + ln * LDSTR;
    #pragma unroll
    for (int kk = 0; kk < 16; ++kk)
        af[kk] = *(const v2f*)(arow + kk * 4 + koff);

    #pragma unroll 1
    for (int nt = 0; nt < 49; ++nt) {
        v8f c = zero;
        // B[k][n] = centersT[n][k]: pair along k contiguous in centersT row n
        const float* bcol = centersT + (size_t)(nt * 16 + ln) * KC + koff;
        #pragma unroll
        for (int kk = 0; kk < 16; ++kk) {
            v2f b = *(const v2f*)(bcol + kk * 4);
            c = __builtin_amdgcn_wmma_f32_16x16x4_f32(false, af[kk], false, b,
                                                      (short)0, c, false, false);
        }
        float* orow = xrec + row0 * DIM + nt * 16 + ln;
        #pragma unroll
        for (int v = 0; v < 8; ++v)
            orow[(size_t)(v + 8 * half) * DIM] = c[v];
    }

    // ================= x0 copy (coalesced float4 block copy of this block's 128 rows)
    const size_t base = (size_t)blockIdx.x * 128 * DIM;    // multiple of 4 floats
    const v4f* src = (const v4f*)(x + base);
    v4f* dst = (v4f*)(xcopy + base);
    const int total = 128 * DIM / 4;                       // 25088
    for (int i = threadIdx.x; i < total; i += 256) dst[i] = src[i];
}

extern "C" void kernel_launch(void* const* d_in, const int* in_sizes, int n_in,
                              void* d_out, int out_size, void* d_ws, size_t ws_size,
                              hipStream_t stream) {
    const float* x       = (const float*)d_in[0];
    const float* centers = (const float*)d_in[1];
    float* out   = (float*)d_out;
    float* xrec  = out;                                    // [N, D]
    float* xcopy = out + (size_t)NROWS * DIM;              // [N, D]

    float* cn2      = (float*)d_ws;                        // 64 floats
    float* centersT = (float*)((char*)d_ws + 256);         // [784, 64] floats

    prep_kernel<<<KC, 256, 0, stream>>>(centers, centersT, cn2);
    cluster_kernel<<<NROWS / 128, 256, 0, stream>>>(x, centers, centersT, cn2, xrec, xcopy);
}